// TensorDense_55937654063718
// MI455X (gfx1250) — compile-verified
//
#include <hip/hip_runtime.h>

// CDNA5 / gfx1250, wave32. All matrix math via V_WMMA_F32_16X16X4_F32.
typedef __attribute__((ext_vector_type(2))) float v2f;
typedef __attribute__((ext_vector_type(4))) float v4f;
typedef __attribute__((ext_vector_type(8))) float v8f;

// Hard scheduling fence: keep pipeline phases from overlapping so register
// live-ranges stay per-phase (peak pressure < 256 VGPRs, no vgpr-msb thrash).
#define SCHED_FENCE() __builtin_amdgcn_sched_barrier(0)

static __device__ __forceinline__ v8f wmma_f32(v2f a, v2f b, v8f c) {
  // (neg_a, A, neg_b, B, c_mod, C, reuse_a, reuse_b)
  return __builtin_amdgcn_wmma_f32_16x16x4_f32(false, a, false, b, (short)0, c,
                                               false, false);
}

// ---------------------------------------------------------------------------
// Real-spherical-harmonic Clebsch-Gordan coefficients, baked offline from the
// reference's _real_cg() construction. Sparse entry pool: (i, j, coeff).
// Path order per output-l matches the reference PATHS scan order exactly:
//   l=0: (0,0) (1,1) (2,2)
//   l=1: (0,1) (1,0) (1,1) (1,2) (2,1) (2,2)
//   l=2: (0,2) (1,1) (1,2) (2,0) (2,1) (2,2)
// ---------------------------------------------------------------------------
static __device__ const unsigned char CG_I[135] = {
  0, 0,1,2, 0,1,2,3,4,
  0, 0, 0,  0, 1, 2,
  1,2, 2,0, 0,1,
  2,0,0,1, 0,1,2, 0,1,2,2,
  0,2,4,1, 1,2,3, 0,3,2,4,
  0,3,1,2,1,4, 0,4,1,3, 0,1,3,2,3,4,
  0,0,0,0,0,
  0,2, 0,1, 0,1,2, 1,2, 2,0,
  1,2,0, 1,2,2,0, 2,0, 1,2,0, 1,2,0,
  0,1,2,3,4,
  4,3,1, 3,2,4,0, 1,3, 1,0,2, 0,1,3,
  0,2,1,3, 0,3,1,2,1,4, 2,0,4,1,3, 3,2,1,0,3,4, 4,2,3,1
};
static __device__ const unsigned char CG_J[135] = {
  0, 0,1,2, 0,1,2,3,4,
  0, 1, 2,  0, 0, 0,
  2,1, 0,2, 1,0,
  0,2,4,1, 1,2,3, 0,3,2,4,
  2,0,0,1, 0,1,2, 0,1,2,2,
  3,0,2,1,4,1, 4,0,3,1, 1,0,2,3,4,3,
  0,1,2,3,4,
  2,0, 1,0, 0,1,2, 2,1, 2,0,
  4,3,1, 3,2,4,0, 1,3, 1,0,2, 0,1,3,
  0,0,0,0,0,
  1,2,0, 1,2,2,0, 2,0, 1,2,0, 1,2,0,
  2,0,3,1, 3,0,2,1,4,1, 2,0,4,1,3, 2,3,0,1,4,3, 2,4,3,1
};
static __device__ const float CG_C[135] = {
  1.0f,
  -0.57735027f,-0.57735027f,-0.57735027f,
  0.44721360f,0.44721360f,0.44721360f,0.44721360f,0.44721360f,
  1.0f,1.0f,1.0f, 1.0f,1.0f,1.0f,
  0.70710678f,-0.70710678f, 0.70710678f,-0.70710678f, 0.70710678f,-0.70710678f,
  0.54772256f,-0.31622777f,-0.54772256f,0.54772256f,
  0.54772256f,0.63245553f,0.54772256f,
  0.54772256f,0.54772256f,-0.31622777f,0.54772256f,
  0.54772256f,-0.31622777f,-0.54772256f,0.54772256f,
  0.54772256f,0.63245553f,0.54772256f,
  0.54772256f,0.54772256f,-0.31622777f,0.54772256f,
  0.31622777f,-0.31622777f,0.54772256f,-0.54772256f,0.31622777f,-0.31622777f,
  0.63245553f,-0.63245553f,0.31622777f,-0.31622777f,
  0.31622777f,-0.31622777f,0.54772256f,-0.54772256f,-0.31622777f,0.31622777f,
  1.0f,1.0f,1.0f,1.0f,1.0f,
  0.70710678f,0.70710678f, 0.70710678f,0.70710678f,
  -0.40824829f,0.81649658f,-0.40824829f,
  0.70710678f,0.70710678f, 0.70710678f,-0.70710678f,
  0.81649658f,-0.40824829f,-0.40824829f,
  0.40824829f,-0.70710678f,0.40824829f,0.40824829f,
  0.70710678f,0.70710678f,
  -0.40824829f,0.40824829f,-0.70710678f,
  -0.81649658f,-0.40824829f,0.40824829f,
  1.0f,1.0f,1.0f,1.0f,1.0f,
  -0.81649658f,0.40824829f,0.40824829f,
  -0.40824829f,0.70710678f,-0.40824829f,-0.40824829f,
  -0.70710678f,-0.70710678f,
  0.40824829f,-0.40824829f,0.70710678f,
  0.81649658f,0.40824829f,-0.40824829f,
  -0.23904572f,-0.23904572f,0.30860670f,0.30860670f,
  0.30860670f,0.30860670f,0.23904572f,0.23904572f,-0.30860670f,-0.30860670f,
  0.53452248f,-0.26726124f,-0.26726124f,0.15430335f,0.15430335f,
  0.23904572f,0.23904572f,0.30860670f,0.30860670f,0.30860670f,0.30860670f,
  -0.23904572f,-0.23904572f,0.46291005f,-0.46291005f
};
static __device__ const short ESTART[3][6][5] = {
  { {0,0,0,0,0},{1,0,0,0,0},{4,0,0,0,0},{0,0,0,0,0},{0,0,0,0,0},{0,0,0,0,0} },
  { {9,10,11,0,0},{12,13,14,0,0},{15,17,19,0,0},
    {21,25,28,0,0},{32,36,39,0,0},{43,49,53,0,0} },
  { {59,60,61,62,63},{64,66,68,71,73},{75,78,82,84,87},
    {90,91,92,93,94},{95,98,102,104,107},{110,114,120,125,131} }
};
static __device__ const unsigned char ECNT[3][6][5] = {
  { {1,0,0,0,0},{3,0,0,0,0},{5,0,0,0,0},{0,0,0,0,0},{0,0,0,0,0},{0,0,0,0,0} },
  { {1,1,1,0,0},{1,1,1,0,0},{2,2,2,0,0},{4,3,4,0,0},{4,3,4,0,0},{6,4,6,0,0} },
  { {1,1,1,1,1},{2,2,3,2,2},{3,4,2,3,3},{1,1,1,1,1},{3,4,2,3,3},{4,6,5,6,4} }
};
static __device__ const unsigned char PL1[3][6] = {
  {0,1,2,0,0,0},{0,1,1,1,2,2},{0,1,1,2,2,2} };
static __device__ const unsigned char PL2[3][6] = {
  {0,1,2,0,0,0},{1,0,1,2,1,2},{2,1,2,0,1,2} };
static __device__ const int AOFF_T[3] = {0, 256, 1024};

// z slab geometry: [kk][n][m_local], padded row for LDS bank spread + 16B align
#define ZROW 132
#define ZK   (16 * ZROW)      // 2112 floats per k-slab
#define ZTOT (3 * ZK)         // up to 3 k's resident

// ---------------------------------------------------------------------------
// Linear1/2:  D[16 mid, 16 samples] = (W^T)[16,64] x X[64,16], per (l, i).
// K-step (t) outer, i inner: one W fragment + 2 X rows in flight per step,
// feeding all (2l+1) accumulators -> bounded register pressure, no movrels.
// ---------------------------------------------------------------------------
template <int L>
__device__ __forceinline__ void linear12(const float* __restrict__ X,
                                         const float* __restrict__ W,
                                         float* __restrict__ dst, int nread,
                                         int nloc, int half) {
  constexpr int ST = 2 * L + 1;
  constexpr int AO = (L == 0) ? 0 : (L == 1 ? 256 : 1024);
  const int K0 = 2 * half;
  const size_t xbase = (size_t)nread * 64u;
  v8f acc[ST];
#pragma unroll
  for (int i = 0; i < ST; ++i)
    acc[i] = (v8f){0.f, 0.f, 0.f, 0.f, 0.f, 0.f, 0.f, 0.f};
#pragma unroll
  for (int t = 0; t < 16; ++t) {
    v2f aw;
    aw.x = W[(4 * t + K0) * 16 + nloc];
    aw.y = W[(4 * t + K0 + 1) * 16 + nloc];
    if constexpr (L == 0) {
      v2f bx = *(const v2f*)&X[xbase + 4 * t + K0];  // contiguous for l=0
      acc[0] = wmma_f32(aw, bx, acc[0]);
    } else {
      float r0[ST], r1[ST];
#pragma unroll
      for (int i = 0; i < ST; ++i) {
        r0[i] = X[(xbase + 4 * t + K0) * ST + i];
        r1[i] = X[(xbase + 4 * t + K0 + 1) * ST + i];
      }
#pragma unroll
      for (int i = 0; i < ST; ++i) {
        v2f bx = {r0[i], r1[i]};
        acc[i] = wmma_f32(aw, bx, acc[i]);
      }
    }
  }
#pragma unroll
  for (int i = 0; i < ST; ++i)
#pragma unroll
    for (int r = 0; r < 8; ++r) {
      int u = r + 8 * half;  // D row owned by this lane/VGPR (ISA 7.12.2)
      dst[AO + (u * 16 + nloc) * ST + i] = acc[i][r];
    }
}

// ---------------------------------------------------------------------------
// Sparse CG tensor product: z[kk][n][m] for one (out-l, path, m-half, k-group).
// v-dimension split in halves to cap live registers (acc[4][8] + bv[8]) while
// Y accumulators stay resident. b128 stores into the padded z slab.
// ---------------------------------------------------------------------------
template <int KCNT>
__device__ __forceinline__ void tp_path(int outl, int p, int klo, int mh,
                                        const float* __restrict__ a_lds,
                                        const float* __restrict__ b_lds,
                                        float* __restrict__ z_lds, int nloc,
                                        int half) {
  const int l1 = PL1[outl][p], l2 = PL2[outl][p];
  const int st1 = 2 * l1 + 1, st2 = 2 * l2 + 1;
  const int ao = AOFF_T[l1] + nloc * st1;
  const int bo = AOFF_T[l2] + nloc * st2;
  const int ubase = mh * 8 + half * 4;  // this lane's 4 u-rows of the m-half
#pragma unroll
  for (int kk = 0; kk < KCNT; ++kk) {
    const int k = klo + kk;
    const int e0 = ESTART[outl][p][k];
    const int ec = ECNT[outl][p][k];
#pragma unroll
    for (int vh = 0; vh < 2; ++vh) {
      float acc[4][8];
#pragma unroll
      for (int ui = 0; ui < 4; ++ui)
#pragma unroll
        for (int v = 0; v < 8; ++v) acc[ui][v] = 0.f;
      for (int e = e0; e < e0 + ec; ++e) {
        const float c = CG_C[e];
        const int jb = bo + CG_J[e] + vh * 8 * 16 * st2;
        float bv[8];
#pragma unroll
        for (int v = 0; v < 8; ++v) bv[v] = b_lds[jb + v * 16 * st2];
#pragma unroll
        for (int ui = 0; ui < 4; ++ui) {
          const float av = a_lds[ao + (ubase + ui) * 16 * st1 + CG_I[e]] * c;
#pragma unroll
          for (int v = 0; v < 8; ++v) acc[ui][v] += av * bv[v];
        }
      }
#pragma unroll
      for (int ui = 0; ui < 4; ++ui) {
        float* zr =
            &z_lds[kk * ZK + nloc * ZROW + (half * 4 + ui) * 16 + vh * 8];
#pragma unroll
        for (int v0 = 0; v0 < 8; v0 += 4) {
          v4f q = {acc[ui][v0], acc[ui][v0 + 1], acc[ui][v0 + 2],
                   acc[ui][v0 + 3]};
          *(v4f*)&zr[v0] = q;  // ds_store_b128
        }
      }
    }
  }
}

// ---------------------------------------------------------------------------
// Linear3: Y[64 w, 16 n] += (W3^T) x z.  Loop nest: path -> m-half -> m0 with
// k innermost so each W3 fragment (8 loads) feeds 4*KCNT WMMAs.
// ---------------------------------------------------------------------------
template <int L, int KLO, int KCNT>
__device__ __forceinline__ void linear3(const float* __restrict__ W3,
                                        float* __restrict__ out,
                                        const float* __restrict__ a_lds,
                                        const float* __restrict__ b_lds,
                                        float* __restrict__ z_lds, int nglob,
                                        int n_total, int nloc, int half) {
  constexpr int ST = 2 * L + 1;
  constexpr int NP = (L == 0) ? 3 : 6;
  constexpr int OFF = (L == 0) ? 0 : (L == 1 ? 64 : 256);
  // fold (1/sqrt(64))^2 * 1/sqrt(tp_mult_l) into the store
  const float s = (L == 0) ? 5.6383672e-4f : 3.9868133e-4f;
  const int K0 = 2 * half;
  v8f y[KCNT][4];
#pragma unroll
  for (int kk = 0; kk < KCNT; ++kk)
#pragma unroll
    for (int t = 0; t < 4; ++t)
      y[kk][t] = (v8f){0.f, 0.f, 0.f, 0.f, 0.f, 0.f, 0.f, 0.f};

  for (int p = 0; p < NP; ++p) {
    const float* Wp = W3 + (size_t)p * 256 * 64;
    __builtin_prefetch(Wp, 0, 1);  // global_prefetch_b8
#pragma unroll
    for (int mh = 0; mh < 2; ++mh) {
      __syncthreads();  // WAR: previous GEMM reads of z complete
      SCHED_FENCE();
      tp_path<KCNT>(L, p, KLO, mh, a_lds, b_lds, z_lds, nloc, half);
      SCHED_FENCE();
      __syncthreads();  // z slab visible across lanes
      const float* Wm = Wp + (size_t)(mh * 128 + K0) * 64 + nloc;
#pragma unroll 2
      for (int m0 = 0; m0 < 128; m0 += 4) {
        const float* Wr = Wm + (size_t)m0 * 64;
        v2f aw0, aw1, aw2, aw3;
        aw0.x = Wr[0];  aw0.y = Wr[64];
        aw1.x = Wr[16]; aw1.y = Wr[80];
        aw2.x = Wr[32]; aw2.y = Wr[96];
        aw3.x = Wr[48]; aw3.y = Wr[112];
#pragma unroll
        for (int kk = 0; kk < KCNT; ++kk) {
          v2f bz = *(const v2f*)&z_lds[kk * ZK + nloc * ZROW + m0 + K0];
          y[kk][0] = wmma_f32(aw0, bz, y[kk][0]);
          y[kk][1] = wmma_f32(aw1, bz, y[kk][1]);
          y[kk][2] = wmma_f32(aw2, bz, y[kk][2]);
          y[kk][3] = wmma_f32(aw3, bz, y[kk][3]);
        }
      }
      SCHED_FENCE();
    }
  }
  if (nglob < n_total) {
#pragma unroll
    for (int kk = 0; kk < KCNT; ++kk) {
      const size_t ob = (size_t)nglob * 576u + OFF + (KLO + kk);
#pragma unroll
      for (int r = 0; r < 8; ++r) {
        const int w = r + 8 * half;
        out[ob + (size_t)w * ST]        = y[kk][0][r] * s;
        out[ob + (size_t)(w + 16) * ST] = y[kk][1][r] * s;
        out[ob + (size_t)(w + 32) * ST] = y[kk][2][r] * s;
        out[ob + (size_t)(w + 48) * ST] = y[kk][3][r] * s;
      }
    }
  }
  SCHED_FENCE();
}

__global__ __launch_bounds__(32)
__attribute__((amdgpu_waves_per_eu(4)))  // pack VGPRs <= 1/4 file (<=256):
                                         // no vgpr-msb, 4 waves/SIMD resident
void tensordense_cdna5(
    const float* __restrict__ x0, const float* __restrict__ x1,
    const float* __restrict__ x2, const float* __restrict__ W10,
    const float* __restrict__ W11, const float* __restrict__ W12,
    const float* __restrict__ W20, const float* __restrict__ W21,
    const float* __restrict__ W22, const float* __restrict__ W30,
    const float* __restrict__ W31, const float* __restrict__ W32,
    float* __restrict__ out, int n_total) {
  __shared__ float a_lds[2304];  // branch-1 activations, all l
  __shared__ float b_lds[2304];  // branch-2 activations, all l
  __shared__ float z_lds[ZTOT];  // z slabs: [<=3 k][16 n][132]
  const int lane = threadIdx.x;
  const int nloc = lane & 15;
  const int half = lane >> 4;
  const int nglob = blockIdx.x * 16 + nloc;
  const int nread = nglob < n_total ? nglob : n_total - 1;  // tail clamp

  linear12<0>(x0, W10, a_lds, nread, nloc, half);
  SCHED_FENCE();
  linear12<1>(x1, W11, a_lds, nread, nloc, half);
  SCHED_FENCE();
  linear12<2>(x2, W12, a_lds, nread, nloc, half);
  SCHED_FENCE();
  linear12<0>(x0, W20, b_lds, nread, nloc, half);
  SCHED_FENCE();
  linear12<1>(x1, W21, b_lds, nread, nloc, half);
  SCHED_FENCE();
  linear12<2>(x2, W22, b_lds, nread, nloc, half);
  SCHED_FENCE();
  __syncthreads();

  linear3<0, 0, 1>(W30, out, a_lds, b_lds, z_lds, nglob, n_total, nloc, half);
  linear3<1, 0, 3>(W31, out, a_lds, b_lds, z_lds, nglob, n_total, nloc, half);
  linear3<2, 0, 3>(W32, out, a_lds, b_lds, z_lds, nglob, n_total, nloc, half);
  linear3<2, 3, 2>(W32, out, a_lds, b_lds, z_lds, nglob, n_total, nloc, half);
}

extern "C" void kernel_launch(void* const* d_in, const int* in_sizes, int n_in,
                              void* d_out, int out_size, void* d_ws,
                              size_t ws_size, hipStream_t stream) {
  (void)n_in; (void)out_size; (void)d_ws; (void)ws_size;
  const float* x0  = (const float*)d_in[0];
  const float* x1  = (const float*)d_in[1];
  const float* x2  = (const float*)d_in[2];
  const float* W10 = (const float*)d_in[3];
  const float* W11 = (const float*)d_in[4];
  const float* W12 = (const float*)d_in[5];
  const float* W20 = (const float*)d_in[6];
  const float* W21 = (const float*)d_in[7];
  const float* W22 = (const float*)d_in[8];
  const float* W30 = (const float*)d_in[9];
  const float* W31 = (const float*)d_in[10];
  const float* W32 = (const float*)d_in[11];
  const int n = in_sizes[0] / 64;          // x0 is [N, 64, 1]
  const int ntiles = (n + 15) / 16;        // 16 samples per wave32 workgroup
  hipLaunchKernelGGL(tensordense_cdna5, dim3(ntiles), dim3(32), 0, stream, x0,
                     x1, x2, W10, W11, W12, W20, W21, W22, W30, W31, W32,
                     (float*)d_out, n);
}